// SAGE_27212912788332
// MI455X (gfx1250) — compile-verified
//
#include <hip/hip_runtime.h>

// ---------------------------------------------------------------------------
// GraphSAGE (3 layers) for MI455X / gfx1250.
//  - sparse aggregate: wave-per-edge float4 gather + global_atomic_add_f32
//    (all operands L2-resident: x/h1/agg ~77MB << 192MB L2)
//  - dense linear layers: V_WMMA_F32_16X16X4_F32 (fp32 precision preserved)
// ---------------------------------------------------------------------------

typedef __attribute__((ext_vector_type(2))) float v2f;
typedef __attribute__((ext_vector_type(8))) float v8f;

__global__ __launch_bounds__(256) void zero_f32(float* __restrict__ p, long n) {
  long i = (long)blockIdx.x * blockDim.x + threadIdx.x;
  long stride = (long)gridDim.x * blockDim.x;
  for (; i < n; i += stride) p[i] = 0.0f;
}

__global__ __launch_bounds__(256) void count_edges(const int* __restrict__ dst,
                                                   float* __restrict__ cnt, int E) {
  int e = blockIdx.x * blockDim.x + threadIdx.x;
  if (e < E) atomicAdd(&cnt[dst[e]], 1.0f);
}

__global__ __launch_bounds__(256) void finalize_deg(float* __restrict__ deg, int N) {
  int i = blockIdx.x * blockDim.x + threadIdx.x;
  if (i < N) deg[i] = 1.0f / fmaxf(deg[i], 1.0f);
}

// One wave per edge: 32 lanes x float4 covers the 128-float feature row.
__global__ __launch_bounds__(256) void scatter_add_rows(const float* __restrict__ X,
                                                        const int* __restrict__ src,
                                                        const int* __restrict__ dst,
                                                        float* __restrict__ AGG, int E) {
  int wave = (int)(((unsigned)blockIdx.x * blockDim.x + threadIdx.x) >> 5);
  int lane = threadIdx.x & 31;
  if (wave >= E) return;
  int s = src[wave];
  int d = dst[wave];
  const float4 v = *(const float4*)(X + (size_t)s * 128 + lane * 4);
  float* o = AGG + (size_t)d * 128 + lane * 4;
  atomicAdd(o + 0, v.x);
  atomicAdd(o + 1, v.y);
  atomicAdd(o + 2, v.z);
  atomicAdd(o + 3, v.w);
}

// Fused SAGE linear: OUT[16 x ncols] = relu?( (AGG*inv_deg)@Wl^T + X@Wr^T + b )
// K fixed at 128. One wave per 16x16 output tile, wave count = ncols/16.
// A-frag (16x4 f32, ISA 7.12.2): lanes 0-15 -> M=lane,K={ks,ks+1};
//                                lanes 16-31 -> M=lane-16,K={ks+2,ks+3}.
// B-frag (4x16) mirrored with N across lanes. C/D: VGPR r -> M=r / M=r+8.
template <bool DUAL, bool RELU>
__global__ __launch_bounds__(256) void sage_gemm(
    const float* __restrict__ AGG, const float* __restrict__ inv_deg,
    const float* __restrict__ X,
    const float* __restrict__ Wl, const float* __restrict__ Wr,
    const float* __restrict__ bias, float* __restrict__ OUT,
    int ncols, int Nrows) {
  __shared__ float xs[16 * 132];                 // pad 128->132 to spread LDS banks
  __shared__ float as[DUAL ? 16 * 132 : 1];

  const int tid  = threadIdx.x;
  const int lane = tid & 31;
  const int wave = tid >> 5;
  const int row0 = blockIdx.x * 16;

  // Stage this block's 16 node rows (and deg-scaled agg rows) into LDS.
  for (int idx = tid; idx < 16 * 128; idx += blockDim.x) {
    int m = idx >> 7, k = idx & 127;
    int r = row0 + m;
    if (r >= Nrows) r = Nrows - 1;               // clamp (rows stay block-local)
    size_t g = (size_t)r * 128 + k;
    xs[m * 132 + k] = X[g];
    if (DUAL) as[m * 132 + k] = AGG[g] * inv_deg[r];
  }
  __syncthreads();

  const int n0   = wave * 16;                    // output-column tile
  const int nlo  = lane & 15;
  const int koff = (lane >> 4) * 2;              // K split across lane halves

  v8f acc;
  const float bv = bias[n0 + nlo];
#pragma unroll
  for (int r = 0; r < 8; ++r) acc[r] = bv;       // seed C with bias

  const float* wr_row = Wr + (size_t)(n0 + nlo) * 128 + koff;
  const float* wl_row = DUAL ? (Wl + (size_t)(n0 + nlo) * 128 + koff) : Wr;
  const float* xa = xs + nlo * 132 + koff;
  const float* aa = DUAL ? (as + nlo * 132 + koff) : xa;

#pragma unroll
  for (int ks = 0; ks < 128; ks += 4) {
    v2f bx = *(const v2f*)(wr_row + ks);
    v2f ax = *(const v2f*)(xa + ks);
    acc = __builtin_amdgcn_wmma_f32_16x16x4_f32(false, ax, false, bx,
                                                (short)0, acc, false, false);
    if (DUAL) {
      v2f bl = *(const v2f*)(wl_row + ks);
      v2f al = *(const v2f*)(aa + ks);
      acc = __builtin_amdgcn_wmma_f32_16x16x4_f32(false, al, false, bl,
                                                  (short)0, acc, false, false);
    }
  }

  const int col = n0 + nlo;
  const int rup = (lane >> 4) << 3;              // +8 rows for upper lane half
#pragma unroll
  for (int r = 0; r < 8; ++r) {
    int row = row0 + rup + r;
    float v = acc[r];
    if (RELU) v = fmaxf(v, 0.0f);
    if (row < Nrows) OUT[(size_t)row * ncols + col] = v;
  }
}

extern "C" void kernel_launch(void* const* d_in, const int* in_sizes, int n_in,
                              void* d_out, int out_size, void* d_ws, size_t ws_size,
                              hipStream_t stream) {
  const float* x   = (const float*)d_in[0];
  const int*   ei  = (const int*)d_in[1];
  const float* W1l = (const float*)d_in[2];
  const float* b1  = (const float*)d_in[3];
  const float* W1r = (const float*)d_in[4];
  const float* W2l = (const float*)d_in[5];
  const float* b2  = (const float*)d_in[6];
  const float* W2r = (const float*)d_in[7];
  const float* W3  = (const float*)d_in[8];
  const float* b3  = (const float*)d_in[9];

  const int N = in_sizes[0] / 128;
  const int E = in_sizes[1] / 2;
  const int* src = ei;
  const int* dst = ei + E;

  // Workspace: inv_deg[Npad] | agg[N*128] | h1[N*128]  (~51 MB)
  float* ws = (float*)d_ws;
  size_t Npad = ((size_t)N + 255) & ~(size_t)255;
  float* inv_deg = ws;
  float* agg = ws + Npad;
  float* h1  = agg + (size_t)N * 128;

  const int nblk16 = (N + 15) / 16;

  // Degrees (counts accumulate into inv_deg, then inverted in place).
  zero_f32<<<2048, 256, 0, stream>>>(inv_deg, (long)Npad + (long)N * 128);
  count_edges<<<(E + 255) / 256, 256, 0, stream>>>(dst, inv_deg, E);
  finalize_deg<<<(N + 255) / 256, 256, 0, stream>>>(inv_deg, N);

  // Layer 1: agg = sum_{src->dst} x[src];  h1 = relu(mean@W1l^T + x@W1r^T + b1)
  scatter_add_rows<<<(E + 7) / 8, 256, 0, stream>>>(x, src, dst, agg, E);
  sage_gemm<true, true><<<nblk16, 256, 0, stream>>>(agg, inv_deg, x, W1l, W1r,
                                                    b1, h1, 128, N);

  // Layer 2: re-aggregate h1; output written in place over agg (block rows are
  // staged into LDS before any global write, and blocks own disjoint rows).
  zero_f32<<<2048, 256, 0, stream>>>(agg, (long)N * 128);
  scatter_add_rows<<<(E + 7) / 8, 256, 0, stream>>>(h1, src, dst, agg, E);
  sage_gemm<true, true><<<nblk16, 256, 0, stream>>>(agg, inv_deg, h1, W2l, W2r,
                                                    b2, agg, 128, N);

  // Output layer: d_out = h2 @ W3^T + b3   (64 cols -> 4 waves per block)
  sage_gemm<false, false><<<nblk16, 128, 0, stream>>>(nullptr, nullptr, agg,
                                                      nullptr, W3, b3,
                                                      (float*)d_out, 64, N);
}